// StandardAttention_35527969472833
// MI455X (gfx1250) — compile-verified
//
#include <hip/hip_runtime.h>
#include <math.h>

typedef float v2f __attribute__((ext_vector_type(2)));
typedef float v8f __attribute__((ext_vector_type(8)));

namespace {
constexpr int Bb = 8, Hh = 32, QL = 16, Dd = 128, PAST = 4080, KV = 4096;
constexpr int NWAVE = 2;
constexpr int KV_PER_WAVE = KV / NWAVE;   // 2048
constexpr int SK = 132;                   // K row stride (floats): conflict-free b64 B-operand reads
constexpr int SV = 136;                   // V row stride (floats): disjoint banks for the two half-waves
constexpr int SP = 20;                    // P row stride (floats): exact 4-bank tiling, 8B aligned pairs
constexpr int KBUF = 16 * SK;             // 2112
constexpr int VBUF = 16 * SV;             // 2176
constexpr int TBUF = KBUF + VBUF;         // one double-buffer slot (K then V)
constexpr int PBUF = 16 * SP;             // 320
constexpr int WSMEM = 2 * TBUF + PBUF;    // 8896 floats / wave (double-buffered K/V)
constexpr int SMEM_TOT = WSMEM * NWAVE;   // 17792 floats = 69.5 KB
}

__global__ __launch_bounds__(NWAVE * 32, 1)
void fa_fwd(const float* __restrict__ qg, const float* __restrict__ kg,
            const float* __restrict__ vg, const float* __restrict__ pkg,
            const float* __restrict__ pvg, float* __restrict__ outg)
{
  __shared__ float smem[SMEM_TOT];
  const int bh   = blockIdx.x;
  const int wave = threadIdx.x >> 5;
  const int lane = threadIdx.x & 31;
  const int n    = lane & 15;   // column-within-tile for B/C layouts, row for A layout
  const int hi   = lane >> 4;   // half-wave select

  float* buf0 = smem + wave * WSMEM;      // [K | V] slot 0
  float* buf1 = buf0 + TBUF;              // [K | V] slot 1
  float* Psm  = buf0 + 2 * TBUF;

  const float scale = 0.08838834764831845f; // 1/sqrt(128)

  const float* qb  = qg  + (size_t)bh * QL * Dd;
  const float* kb  = kg  + (size_t)bh * QL * Dd;
  const float* vb  = vg  + (size_t)bh * QL * Dd;
  const float* pkb = pkg + (size_t)bh * PAST * Dd;
  const float* pvb = pvg + (size_t)bh * PAST * Dd;

  // Issue one 16x128 K tile + 16x128 V tile as 32 async wave-wide b128 copies
  // (each instruction moves 512B straight into LDS, tracked by ASYNCcnt).
  auto issue_tile = [&](int j0, float* Kb, float* Vb) {
#pragma unroll
    for (int r = 0; r < 16; ++r) {
      const int j = j0 + r;
      const float* krow = (j < PAST) ? (pkb + (size_t)j * Dd) : (kb + (size_t)(j - PAST) * Dd);
      const float* vrow = (j < PAST) ? (pvb + (size_t)j * Dd) : (vb + (size_t)(j - PAST) * Dd);
      const unsigned kl = (unsigned)(uintptr_t)&Kb[r * SK + 4 * lane]; // generic[31:0] == LDS offset
      const unsigned vl = (unsigned)(uintptr_t)&Vb[r * SV + 4 * lane];
      asm volatile("global_load_async_to_lds_b128 %0, %1, off"
                   :: "v"(kl), "v"(krow + 4 * lane) : "memory");
      asm volatile("global_load_async_to_lds_b128 %0, %1, off"
                   :: "v"(vl), "v"(vrow + 4 * lane) : "memory");
    }
  };

  // ---- Q (16x128) into registers, WMMA A-layout, pre-scaled ----
  // A 16x4 f32 layout: lanes 0-15 hold K=0,1 in v0,v1; lanes 16-31 hold K=2,3.
  v2f qa[32];
#pragma unroll
  for (int kc = 0; kc < 32; ++kc) {
    const float* p = qb + n * Dd + 4 * kc + 2 * hi;
    v2f t; t.x = p[0] * scale; t.y = p[1] * scale;
    qa[kc] = t;
  }

  v8f acc[8];
#pragma unroll
  for (int c = 0; c < 8; ++c) acc[c] = (v8f){0,0,0,0,0,0,0,0};
  float m_r[8], l_r[8];
#pragma unroll
  for (int r = 0; r < 8; ++r) { m_r[r] = -3.4e38f; l_r[r] = 0.f; }

  const int j_begin = wave * KV_PER_WAVE;
  const int NT = KV_PER_WAVE / 16;

  issue_tile(j_begin, buf0, buf0 + KBUF);     // prologue prefetch into slot 0

  for (int t = 0; t < NT; ++t) {
    const int j0 = j_begin + 16 * t;
    float* Kc = ((t & 1) ? buf1 : buf0);
    float* Vc = Kc + KBUF;

    if (t + 1 < NT) {
      // WAR fence: async writes are unordered vs the DS pipe, so make sure the
      // previous iteration's DS reads of the target slot have drained.
      asm volatile("s_wait_dscnt 0x0" ::: "memory");
      float* Kn = ((t & 1) ? buf0 : buf1);
      issue_tile(j0 + 16, Kn, Kn + KBUF);
      // 32 newer ops outstanding; async loads complete in order, so <=32 means
      // the current tile's 32 copies have landed.
      asm volatile("s_wait_asynccnt 0x20" ::: "memory");
    } else {
      asm volatile("s_wait_asynccnt 0x0" ::: "memory");
    }

    // ---- S(16x16) = (Q*scale) . K^T : 32 chained V_WMMA_F32_16X16X4_F32 ----
    v8f s = (v8f){0,0,0,0,0,0,0,0};
#pragma unroll
    for (int kc = 0; kc < 32; ++kc) {
      // B 4x16 layout: v0 = K_blk[n][4kc+2hi], v1 = K_blk[n][4kc+2hi+1]
      v2f bk = *(const v2f*)&Kc[n * SK + 4 * kc + 2 * hi];
      s = __builtin_amdgcn_wmma_f32_16x16x4_f32(false, qa[kc], false, bk,
                                                (short)0, s, false, false);
    }

    // ---- causal mask: only the single boundary tile triggers this (uniform) ----
    if (j0 + 15 > PAST) {
#pragma unroll
      for (int r = 0; r < 8; ++r) {
        const int qrow = r + 8 * hi;               // C layout: VGPR r holds rows r / r+8
        if (j0 + n > PAST + qrow) s[r] = -3.4e38f; // j > (kv_len - q_len) + i
      }
    }

    // ---- online softmax: per-row stats, rows live across 16-lane halves ----
    float corr[8];
#pragma unroll
    for (int r = 0; r < 8; ++r) {
      float x = s[r];
      x = fmaxf(x, __shfl_xor(x, 1, 32));
      x = fmaxf(x, __shfl_xor(x, 2, 32));
      x = fmaxf(x, __shfl_xor(x, 4, 32));
      x = fmaxf(x, __shfl_xor(x, 8, 32));
      const float mnew = fmaxf(m_r[r], x);
      corr[r] = __expf(m_r[r] - mnew);
      const float pval = __expf(s[r] - mnew);
      Psm[(r + 8 * hi) * SP + n] = pval;           // C layout -> row-major P tile
      float rs = pval;
      rs += __shfl_xor(rs, 1, 32);
      rs += __shfl_xor(rs, 2, 32);
      rs += __shfl_xor(rs, 4, 32);
      rs += __shfl_xor(rs, 8, 32);
      l_r[r] = l_r[r] * corr[r] + rs;
      m_r[r] = mnew;
    }
#pragma unroll
    for (int c = 0; c < 8; ++c)
#pragma unroll
      for (int r = 0; r < 8; ++r) acc[c][r] *= corr[r];

    // ---- P tile back out of LDS in A-layout ----
    v2f pa[4];
#pragma unroll
    for (int kc = 0; kc < 4; ++kc)
      pa[kc] = *(const v2f*)&Psm[n * SP + 4 * kc + 2 * hi];

    // ---- acc(16x128) += P(16x16) . V(16x128) : 32 WMMAs ----
#pragma unroll
    for (int c = 0; c < 8; ++c) {
#pragma unroll
      for (int kc = 0; kc < 4; ++kc) {
        v2f bv;
        bv.x = Vc[(4 * kc + 2 * hi)     * SV + c * 16 + n];
        bv.y = Vc[(4 * kc + 2 * hi + 1) * SV + c * 16 + n];
        acc[c] = __builtin_amdgcn_wmma_f32_16x16x4_f32(false, pa[kc], false, bv,
                                                       (short)0, acc[c], false, false);
      }
    }
  }

  // -------- cross-wave flash-decoding merge (LDS reused after barrier) --------
  __syncthreads();
  float* accS = smem;                       // [wave][16][128]
  float* stat = smem + NWAVE * QL * Dd;     // [wave][32]: m[16] then l[16]
#pragma unroll
  for (int c = 0; c < 8; ++c)
#pragma unroll
    for (int r = 0; r < 8; ++r)
      accS[wave * (QL * Dd) + (r + 8 * hi) * Dd + c * 16 + n] = acc[c][r];
  if (n == 0) {
#pragma unroll
    for (int r = 0; r < 8; ++r) {
      stat[wave * 32 + (r + 8 * hi)]      = m_r[r];
      stat[wave * 32 + 16 + (r + 8 * hi)] = l_r[r];
    }
  }
  __syncthreads();

  float* ob = outg + (size_t)bh * QL * Dd;
#pragma unroll
  for (int cc = 0; cc < 8 / NWAVE; ++cc) {
    const int c = (8 / NWAVE) * wave + cc;
#pragma unroll
    for (int r = 0; r < 8; ++r) {
      const int qrow = r + 8 * hi;
      const int d = c * 16 + n;
      float M = -3.4e38f;
#pragma unroll
      for (int w = 0; w < NWAVE; ++w) M = fmaxf(M, stat[w * 32 + qrow]);
      float Ls = 0.f, o = 0.f;
#pragma unroll
      for (int w = 0; w < NWAVE; ++w) {
        const float e = __expf(stat[w * 32 + qrow] - M);
        Ls += e * stat[w * 32 + 16 + qrow];
        o  += e * accS[w * (QL * Dd) + qrow * Dd + d];
      }
      ob[qrow * Dd + d] = o / Ls;
    }
  }
}

extern "C" void kernel_launch(void* const* d_in, const int* in_sizes, int n_in,
                              void* d_out, int out_size, void* d_ws, size_t ws_size,
                              hipStream_t stream) {
  (void)in_sizes; (void)n_in; (void)d_ws; (void)ws_size; (void)out_size;
  const float* q  = (const float*)d_in[0];
  const float* k  = (const float*)d_in[1];
  const float* v  = (const float*)d_in[2];
  const float* pk = (const float*)d_in[3];
  const float* pv = (const float*)d_in[4];
  float* out = (float*)d_out;
  dim3 grid(Bb * Hh);        // 256 (b,h) workgroups
  dim3 block(NWAVE * 32);    // 2 wave32 per workgroup, kv split across waves
  fa_fwd<<<grid, block, 0, stream>>>(q, k, v, pk, pv, out);
}